// WLLoss_test_28673201668770
// MI455X (gfx1250) — compile-verified
//
#include <hip/hip_runtime.h>
#include <math.h>

typedef __attribute__((ext_vector_type(2))) float v2f;
typedef __attribute__((ext_vector_type(8))) float v8f;

// ---------------- workspace layout (floats) ----------------
// [0, 2240)                : W matrix, 112 rows x 20 cols, rows >=100 are zero
// per scale s (s=0..2), base = STATE_OFF + s*STATE_STRIDE:
//   [0..10] state: 0 n_pos, 1 loss_pos, 2 count_neg, 3 sum_ce_ttm,
//                  4 sum_ce_neg, 5 sum_negm, 6 sum_ttm_perpix,
//                  7 n_neg(float), 8 remaining(uint bits), 9 selected_sum,
//                  10 prefix(uint bits)
//   [16, 16+2048)      : histogram counts (uint)
//   [16+2048, 16+4096) : histogram value sums (float)
#define W_COLS 20
#define W_OFF 0
#define STATE_OFF 2304
#define HIST_N 2048
#define STATE_STRIDE (16 + 2 * HIST_N)
#define WS_FLOATS (STATE_OFF + 3 * STATE_STRIDE)

__device__ __constant__ float DEC_LO[10] = {
    0.027333068345077982f,  0.029519490925774643f, -0.039134249302383094f,
    0.1993975339773936f,    0.7234076904024206f,    0.6339789634582119f,
    0.01660210576452232f,  -0.17532808990845047f,  -0.021101834024758855f,
    0.019538882735286728f};

__device__ inline float lse2(float a, float b) {
  return fmaxf(a, b) + log1pf(expf(-fabsf(a - b)));
}
__device__ inline float smooth_l1(float d) {
  float ad = fabsf(d);
  return ad < 1.f ? 0.5f * d * d : ad - 0.5f;
}

// One stage of the inverse DWT used by the reference:
// upsample-by-2 (lhs_dilation), pad 1/1, correlate with 10-tap kernel.
// input length n -> output length 2n-8.
__device__ void idwt_up(const float* in, int n, float* out) {
  int on = 2 * n - 8;
  for (int o = 0; o < on; ++o) {
    float s = 0.f;
    for (int k = (o & 1) ? 0 : 1; k < 10; k += 2) {  // need (o+k) odd
      int idx = (o + k - 1) >> 1;
      if (idx >= 0 && idx < n) s += DEC_LO[k] * in[idx];
    }
    out[o] = s;
  }
}

// Zero workspace; build the composed 100x20 reconstruction matrix W by
// pushing each of the 20 basis vectors through the three IDWT stages
// (20 -> 32, crop 31 -> 54 -> 100). Rows 100..111 stay zero (WMMA padding).
__global__ void __launch_bounds__(256) k_init(float* ws) {
  int tid = threadIdx.x;
  for (int i = tid; i < WS_FLOATS; i += blockDim.x) ws[i] = 0.f;
  __syncthreads();
  if (tid < W_COLS) {
    float c0[20];
    for (int i = 0; i < 20; ++i) c0[i] = 0.f;
    c0[tid] = 1.f;
    float r1[32], r2[54], r3[100];
    idwt_up(c0, 20, r1);
    idwt_up(r1, 31, r2);  // crop to 31: only first 31 entries used
    idwt_up(r2, 54, r3);
    for (int m = 0; m < 100; ++m) ws[W_OFF + m * W_COLS + tid] = r3[m];
  }
}

// Per-pixel CE / mask reductions (6 scalar accumulators).
__global__ void __launch_bounds__(256) k_scalar(const float* __restrict__ cls,
                                                const float* __restrict__ gt,
                                                int P, int HH, float* st) {
  int lane = threadIdx.x & 31;
  float a0 = 0, a1 = 0, a2 = 0, a3 = 0, a4 = 0, a5 = 0;
  for (int p = blockIdx.x * blockDim.x + threadIdx.x; p < P;
       p += gridDim.x * blockDim.x) {
    int n = p / HH, rem = p - n * HH;
    const float* c = cls + (n * 4) * HH + rem;
    const float* g = gt + (n * 45) * HH + rem;
    float t0 = c[0], t1 = c[HH], q0 = c[2 * HH], q1 = c[3 * HH];
    float trm = g[0], tclm = g[HH], trn = g[2 * HH];
    float ce_tr = lse2(t0, t1) - (trm > 0.f ? t1 : t0);
    bool pos = (trm > 0.f) && (trn > 0.f);
    bool neg = (trm <= 0.f) && (trn > 0.f);
    float ce_tcl = lse2(q0, q1) - (tclm > 0.f ? q1 : q0);
    float ttm = trm * trn, negm = 1.f - ttm;
    a0 += pos ? 1.f : 0.f;
    a1 += pos ? ce_tr : 0.f;
    a2 += neg ? 1.f : 0.f;
    a3 += ce_tcl * ttm;
    a4 += ce_tcl * negm;
    a5 += negm;
  }
#pragma unroll
  for (int off = 16; off; off >>= 1) {
    a0 += __shfl_down(a0, off, 32);
    a1 += __shfl_down(a1, off, 32);
    a2 += __shfl_down(a2, off, 32);
    a3 += __shfl_down(a3, off, 32);
    a4 += __shfl_down(a4, off, 32);
    a5 += __shfl_down(a5, off, 32);
  }
  if (lane == 0) {
    atomicAdd(&st[0], a0);
    atomicAdd(&st[1], a1);
    atomicAdd(&st[2], a2);
    atomicAdd(&st[3], a3);
    atomicAdd(&st[4], a4);
    atomicAdd(&st[5], a5);
  }
}

// Contour loss via FP32 WMMA: each wave processes 16 pixels per iteration.
// D[112x16] = W[112x20] @ d[20x16]; smooth_l1(D + dc) summed over valid
// rows (<100) and x/y, then masked by ttm and atomically accumulated.
// Tail handling is branchless: clamp pixel index, multiply by a 0/1 mask,
// so every load is unconditional (no exec-diamonds, no spill pressure).
__global__ void __launch_bounds__(256) k_contour(const float* __restrict__ reg,
                                                 const float* __restrict__ gt,
                                                 int P, int HH,
                                                 const float* __restrict__ Wm,
                                                 float* st) {
  int lane = threadIdx.x & 31;
  int col = lane & 15, half = lane >> 4;

  // A fragments (16x4 f32 layout): lanes 0-15 hold K=0,1; lanes 16-31 K=2,3.
  v2f A[7][5];
#pragma unroll
  for (int t = 0; t < 7; ++t)
#pragma unroll
    for (int kc = 0; kc < 5; ++kc) {
      int m = t * 16 + col, k = kc * 4 + half * 2;
      A[t][kc][0] = Wm[m * W_COLS + k];
      A[t][kc][1] = Wm[m * W_COLS + k + 1];
    }

  int G = (P + 15) >> 4;
  int wave = blockIdx.x * (blockDim.x >> 5) + (threadIdx.x >> 5);
  int wstep = gridDim.x * (blockDim.x >> 5);
  for (int g = wave; g < G; g += wstep) {
    int pix = g * 16 + col;
    float vmask = (pix < P) ? 1.f : 0.f;
    int pixc = (pix < P) ? pix : (P - 1);  // clamped: loads always in-bounds
    int n = pixc / HH, rem = pixc - n * HH;
    const float* rp = reg + (n * 42) * HH + rem;
    const float* gp = gt + (n * 45) * HH + rem;
    float dcx = vmask * (rp[20 * HH] - gp[23 * HH]);
    float dcy = vmask * (rp[41 * HH] - gp[44 * HH]);
    float ttm = vmask * gp[2 * HH] * gp[0];

    // B fragments (4x16 f32, mirrors A): per-pixel coefficient differences.
    v2f Bx[5], By[5];
#pragma unroll
    for (int kc = 0; kc < 5; ++kc) {
      int k = kc * 4 + half * 2;
      Bx[kc][0] = vmask * (rp[k * HH] - gp[(3 + k) * HH]);
      Bx[kc][1] = vmask * (rp[(k + 1) * HH] - gp[(4 + k) * HH]);
      By[kc][0] = vmask * (rp[(21 + k) * HH] - gp[(24 + k) * HH]);
      By[kc][1] = vmask * (rp[(22 + k) * HH] - gp[(25 + k) * HH]);
    }

    float partial = 0.f;
#pragma unroll
    for (int t = 0; t < 7; ++t) {
      v8f cx = {};
      v8f cy = {};
#pragma unroll
      for (int kc = 0; kc < 5; ++kc) {
        cx = __builtin_amdgcn_wmma_f32_16x16x4_f32(false, A[t][kc], false,
                                                   Bx[kc], (short)0, cx,
                                                   false, false);
        cy = __builtin_amdgcn_wmma_f32_16x16x4_f32(false, A[t][kc], false,
                                                   By[kc], (short)0, cy,
                                                   false, false);
      }
#pragma unroll
      for (int r = 0; r < 8; ++r) {
        int Mrow = t * 16 + half * 8 + r;
        float keep = (Mrow < 100) ? 1.f : 0.f;
        partial += keep * (smooth_l1(cx[r] + dcx) + smooth_l1(cy[r] + dcy));
      }
    }
    // combine the two half-wave contributions for each pixel
    partial += __shfl_xor(partial, 16, 32);
    float contrib = (half == 0) ? ttm * partial : 0.f;
#pragma unroll
    for (int off = 16; off; off >>= 1) contrib += __shfl_down(contrib, off, 32);
    if (lane == 0) atomicAdd(&st[6], contrib);
  }
}

// Compute n_neg and init radix-select state.
__global__ void k_nneg(float* ws) {
  int s = threadIdx.x;
  if (s >= 3 || blockIdx.x) return;
  float* st = ws + STATE_OFF + s * STATE_STRIDE;
  unsigned* stu = (unsigned*)st;
  float np = st[0], cn = st[2];
  float nn = (np > 0.f) ? fminf(cn, floorf(3.f * np)) : 100.f;
  st[7] = nn;
  stu[8] = (unsigned)nn;  // remaining
  st[9] = 0.f;            // selected sum
  stu[10] = 0u;           // prefix
}

// Radix-select histogram pass over negatives' CE (bit-monotonic: CE >= 0).
__global__ void __launch_bounds__(256) k_hist(const float* __restrict__ cls,
                                              const float* __restrict__ gt,
                                              int P, int HH, float* st,
                                              int shift, int prevShift,
                                              unsigned binMask) {
  unsigned* stu = (unsigned*)st;
  unsigned remaining = stu[8];
  if (remaining == 0) return;
  unsigned prefix = stu[10];
  unsigned* cnt = (unsigned*)(st + 16);
  float* sums = st + 16 + HIST_N;
  __shared__ unsigned scnt[HIST_N];
  __shared__ float ssum[HIST_N];
  for (int i = threadIdx.x; i < HIST_N; i += blockDim.x) {
    scnt[i] = 0u;
    ssum[i] = 0.f;
  }
  __syncthreads();
  for (int p = blockIdx.x * blockDim.x + threadIdx.x; p < P;
       p += gridDim.x * blockDim.x) {
    int n = p / HH, rem = p - n * HH;
    const float* g = gt + (n * 45) * HH + rem;
    float trm = g[0], trn = g[2 * HH];
    if (!(trm <= 0.f && trn > 0.f)) continue;  // negatives only
    const float* c = cls + (n * 4) * HH + rem;
    float t0 = c[0], t1 = c[HH];
    float ce = lse2(t0, t1) - t0;  // target == 0 for negatives
    unsigned u = __float_as_uint(ce);
    if (prevShift < 32 && (u >> prevShift) != (prefix >> prevShift)) continue;
    unsigned bin = (u >> shift) & binMask;
    atomicAdd(&scnt[bin], 1u);
    atomicAdd(&ssum[bin], ce);
  }
  __syncthreads();
  for (int i = threadIdx.x; i < HIST_N; i += blockDim.x) {
    if (scnt[i]) {
      atomicAdd(&cnt[i], scnt[i]);
      atomicAdd(&sums[i], ssum[i]);
    }
  }
}

// Walk histogram high->low; final pass resolves exact value for ties.
__global__ void k_scan(float* ws, int shift, int last) {
  int s = threadIdx.x;
  if (s >= 3 || blockIdx.x) return;
  float* st = ws + STATE_OFF + s * STATE_STRIDE;
  unsigned* stu = (unsigned*)st;
  unsigned* cnt = (unsigned*)(st + 16);
  float* sums = st + 16 + HIST_N;
  unsigned remaining = stu[8];
  float sel = st[9];
  unsigned prefix = stu[10];
  if (remaining) {
    for (int b = HIST_N - 1; b >= 0; --b) {
      unsigned c = cnt[b];
      if (c == 0) continue;
      if (c <= remaining) {
        sel += sums[b];
        remaining -= c;
        if (!remaining) break;
      } else {
        if (last) {
          float v = __uint_as_float(prefix | ((unsigned)b << shift));
          sel += (float)remaining * v;
          remaining = 0;
        } else {
          prefix |= (unsigned)b << shift;
        }
        break;
      }
    }
  }
  stu[8] = remaining;
  st[9] = sel;
  stu[10] = prefix;
  for (int b = 0; b < HIST_N; ++b) {
    cnt[b] = 0u;
    sums[b] = 0.f;
  }
}

__global__ void k_final(const float* __restrict__ ws, float* out) {
  if (threadIdx.x || blockIdx.x) return;
  float L0 = 0, L1 = 0, L2 = 0;
  for (int s = 0; s < 3; ++s) {
    const float* st = ws + STATE_OFF + s * STATE_STRIDE;
    float np = st[0], nn = st[7];
    float lp = (np > 0.f) ? st[1] : 0.f;
    L0 += (lp + st[9]) / (np + nn);
    float mp = st[3] / fmaxf(np, 1.f);
    float mn = st[4] / fmaxf(st[5], 1.f);
    L1 += (np > 0.f) ? (mp + 0.5f * mn) : 0.f;
    L2 += (np > 0.f) ? (0.5f * st[6] / (np * 200.f)) : 0.f;
  }
  out[0] = L0;
  out[1] = L1;
  out[2] = L2;
}

extern "C" void kernel_launch(void* const* d_in, const int* in_sizes, int n_in,
                              void* d_out, int out_size, void* d_ws,
                              size_t ws_size, hipStream_t stream) {
  (void)in_sizes; (void)n_in; (void)out_size; (void)ws_size;
  float* ws = (float*)d_ws;
  const int Hs[3] = {100, 50, 25};
  const int Nb = 8;

  k_init<<<1, 256, 0, stream>>>(ws);

  for (int s = 0; s < 3; ++s) {
    const float* cls = (const float*)d_in[s * 3 + 0];
    const float* reg = (const float*)d_in[s * 3 + 1];
    const float* gtp = (const float*)d_in[s * 3 + 2];
    int HH = Hs[s] * Hs[s], P = Nb * HH;
    float* st = ws + STATE_OFF + s * STATE_STRIDE;

    int blocksS = (P + 255) / 256;
    if (blocksS > 512) blocksS = 512;
    k_scalar<<<blocksS, 256, 0, stream>>>(cls, gtp, P, HH, st);

    int G = (P + 15) / 16;
    int blocksC = (G + 7) / 8;
    if (blocksC > 1024) blocksC = 1024;
    k_contour<<<blocksC, 256, 0, stream>>>(reg, gtp, P, HH, ws + W_OFF, st);
  }

  k_nneg<<<1, 32, 0, stream>>>(ws);

  const int shifts[3] = {21, 10, 0};
  const int prevs[3] = {32, 21, 10};
  const unsigned masks[3] = {2047u, 2047u, 1023u};
  for (int pass = 0; pass < 3; ++pass) {
    for (int s = 0; s < 3; ++s) {
      const float* cls = (const float*)d_in[s * 3 + 0];
      const float* gtp = (const float*)d_in[s * 3 + 2];
      int HH = Hs[s] * Hs[s], P = Nb * HH;
      float* st = ws + STATE_OFF + s * STATE_STRIDE;
      int blocksH = (P + 255) / 256;
      if (blocksH > 256) blocksH = 256;
      k_hist<<<blocksH, 256, 0, stream>>>(cls, gtp, P, HH, st, shifts[pass],
                                          prevs[pass], masks[pass]);
    }
    k_scan<<<1, 32, 0, stream>>>(ws, shifts[pass], pass == 2 ? 1 : 0);
  }

  k_final<<<1, 32, 0, stream>>>(ws, (float*)d_out);
}